// criss2D_66254165508509
// MI455X (gfx1250) — compile-verified
//
#include <hip/hip_runtime.h>
#include <math.h>

// CDNA5 (gfx1250) wave32 WMMA types
typedef __attribute__((ext_vector_type(2))) float v2f;
typedef __attribute__((ext_vector_type(8))) float v8f;

#define D_ 24
#define C_ 16
#define H_ 96
#define W_ 192
#define NEG_INF_ (-1.0e9f)

// ---------------------------------------------------------------------------
// Async global->LDS staging (GLOBAL_LOAD_ASYNC_TO_LDS_*, tracked by ASYNCcnt).
// Builtin signatures (from hipcc diagnostics):
//   b128: (int4 addrspace(1)*, int4 addrspace(3)*, imm offset, imm cpol)
//   b32 : (int  addrspace(1)*, int  addrspace(3)*, imm offset, imm cpol)
// ---------------------------------------------------------------------------
#if defined(__HIP_DEVICE_COMPILE__) &&                                        \
    __has_builtin(__builtin_amdgcn_global_load_async_to_lds_b128) &&          \
    __has_builtin(__builtin_amdgcn_global_load_async_to_lds_b32)
#define HAVE_ASYNC_LDS 1
#endif

typedef int i32x4 __attribute__((__vector_size__(16)));
typedef __attribute__((address_space(1))) i32x4 g_i32x4;
typedef __attribute__((address_space(3))) i32x4 l_i32x4;
typedef __attribute__((address_space(1))) int   g_i32;
typedef __attribute__((address_space(3))) int   l_i32;

__device__ __forceinline__ void async_ld_b128(float* lds, const float* g) {
#ifdef HAVE_ASYNC_LDS
    __builtin_amdgcn_global_load_async_to_lds_b128(
        (g_i32x4*)(void*)g, (l_i32x4*)(void*)lds, 0, 0);
#else
    lds[0] = g[0]; lds[1] = g[1]; lds[2] = g[2]; lds[3] = g[3];
#endif
}

__device__ __forceinline__ void async_ld_b32(float* lds, const float* g) {
#ifdef HAVE_ASYNC_LDS
    __builtin_amdgcn_global_load_async_to_lds_b32(
        (g_i32*)(void*)g, (l_i32*)(void*)lds, 0, 0);
#else
    lds[0] = g[0];
#endif
}

__device__ __forceinline__ void wait_async_lds() {
#ifdef HAVE_ASYNC_LDS
#if __has_builtin(__builtin_amdgcn_s_wait_asynccnt)
    __builtin_amdgcn_s_wait_asynccnt(0);
#else
    asm volatile("s_wait_asynccnt 0x0" ::: "memory");
#endif
#endif
}

// ---------------------------------------------------------------------------
// Projection kernel: q = q_w@x + q_b (2ch), k likewise, v = v_w@x + v_b (16ch).
// One wave per (b, h, 16-wide w tile). v projection uses V_WMMA_F32_16X16X4_F32:
//   M = output channel (16), N = spatial (16), K = input channel (16, 4 steps).
// ---------------------------------------------------------------------------
__global__ __launch_bounds__(32) void cca_proj_kernel(
    const float* __restrict__ x,
    const float* __restrict__ qw, const float* __restrict__ qbias,
    const float* __restrict__ kw, const float* __restrict__ kbias,
    const float* __restrict__ vw, const float* __restrict__ vbias,
    float* __restrict__ q, float* __restrict__ k, float* __restrict__ v,
    int in_cdhw)
{
    const int w0  = blockIdx.x * 16;
    const int h   = blockIdx.y;
    const int b   = blockIdx.z;
    const int tid = threadIdx.x;
    const int m   = tid & 15;     // lane within half-wave
    const int hi  = tid >> 4;     // half-wave select

    // stride 20 floats: rows 16B-aligned (80B) and bank step 20 -> 16 distinct
    __shared__ float xt[16][20];   // [in_ch][n] x tile
    __shared__ float vwl[16][20];  // [out_ch][in_ch] v weights

    // 64 B128 chunks each: 2 per lane, async into LDS
    for (int i = tid; i < 64; i += 32) {          // x tile: 16 rows x 4 chunks
        int c = i >> 2, ch = (i & 3) * 4;
        int idx = in_cdhw ? ((c * D_ + b) * H_ + h) * W_ + (w0 + ch)
                          : ((b * C_ + c) * H_ + h) * W_ + (w0 + ch);
        async_ld_b128(&xt[c][ch], &x[idx]);
    }
    for (int i = tid; i < 64; i += 32) {          // v weights: 16 rows x 4 chunks
        int c = i >> 2, ch = (i & 3) * 4;
        async_ld_b128(&vwl[c][ch], &vw[c * 16 + ch]);
    }
    wait_async_lds();
    __syncthreads();

    // v projection: D[mo][n] = sum_kc vw[mo][kc] * x[kc][n]
    v8f acc = {0.f, 0.f, 0.f, 0.f, 0.f, 0.f, 0.f, 0.f};
#pragma unroll
    for (int kb4 = 0; kb4 < 16; kb4 += 4) {
        v2f a, bm;
        // A 16x4 f32 layout: VGPR0 = K(kb4 + 0|2), VGPR1 = K(kb4 + 1|3); row M = lane&15
        a[0]  = vwl[m][kb4 + 2 * hi];
        a[1]  = vwl[m][kb4 + 2 * hi + 1];
        // B 4x16 f32 layout (mirrored): B[k][n], n = lane&15
        bm[0] = xt[kb4 + 2 * hi][m];
        bm[1] = xt[kb4 + 2 * hi + 1][m];
        acc = __builtin_amdgcn_wmma_f32_16x16x4_f32(
            false, a, false, bm, (short)0, acc, false, false);
    }
#pragma unroll
    for (int r = 0; r < 8; ++r) {
        int co = r + 8 * hi;   // D layout: VGPR r -> row r (lanes 0-15) / r+8 (lanes 16-31)
        v[((b * C_ + co) * H_ + h) * W_ + (w0 + m)] = acc[r] + vbias[co];
    }

    // q,k projections (2 output channels): lane = co*16 + n
    {
        int co = hi, n = m;
        float sq = qbias[co], sk = kbias[co];
#pragma unroll
        for (int c = 0; c < 16; ++c) {
            float xv = xt[c][n];
            sq += qw[co * 16 + c] * xv;
            sk += kw[co * 16 + c] * xv;
        }
        q[((b * 2 + co) * H_ + h) * W_ + (w0 + n)] = sq;
        k[((b * 2 + co) * H_ + h) * W_ + (w0 + n)] = sk;
    }
}

// ---------------------------------------------------------------------------
// Fused criss-cross attention: one wave per (b, w, 16-high h tile).
// Async-LDS staging -> energies -> joint softmax(96+192) -> out_H via 24x
// chained f32 WMMA, out_W via LDS-staged batched matvec, residual + gamma.
// ---------------------------------------------------------------------------
__global__ __launch_bounds__(32) void cca_attend_kernel(
    const float* __restrict__ x,
    const float* __restrict__ q, const float* __restrict__ k,
    const float* __restrict__ v, const float* __restrict__ gamma,
    float* __restrict__ out, int in_cdhw, int out_cdhw)
{
    const int w   = blockIdx.x;        // 0..191
    const int h0  = blockIdx.y * 16;   // 6 tiles of 16
    const int b   = blockIdx.z;        // 0..23
    const int tid = threadIdx.x;
    const int m   = tid & 15;
    const int hi  = tid >> 4;

    __shared__ float kcol[2][96];       // k[:, j, w], j over H (strided gather)
    __shared__ float qt[2][16];         // q[:, h0+hh, w]
    __shared__ float krow[2][16][192];  // k[:, h0+hh, j], contiguous rows
    __shared__ float attH[16][97];      // odd stride -> conflict free
    __shared__ float attW[16][193];
    __shared__ float vcol[16][97];      // v[c, j, w], WMMA B source (strided gather)
    __shared__ float vr[16][196];       // staged v row; stride 196: 16B-aligned rows,
                                        // bank step 4 -> conflict free across c

    // ---- async staging: strided gathers as B32, contiguous rows as B128 ----
    for (int i = tid; i < 2 * 96; i += 32)        // kcol
        async_ld_b32(&kcol[0][0] + i, &k[((b * 2 + i / 96) * H_ + i % 96) * W_ + w]);
    {
        int c2 = hi, hh = m;                      // qt (one element per lane)
        async_ld_b32(&qt[c2][hh], &q[((b * 2 + c2) * H_ + h0 + hh) * W_ + w]);
    }
    for (int i = tid; i < 2 * 16 * 48; i += 32) { // krow: 32 rows x 48 B128 chunks
        int c2 = i / (16 * 48), rem = i % (16 * 48);
        int hh = rem / 48, j = (rem % 48) * 4;
        async_ld_b128(&krow[c2][hh][j], &k[((b * 2 + c2) * H_ + h0 + hh) * W_ + j]);
    }
    for (int i = tid; i < 16 * 96; i += 32) {     // vcol (strided by W)
        int c = i / 96, j = i % 96;
        async_ld_b32(&vcol[c][j], &v[((b * C_ + c) * H_ + j) * W_ + w]);
    }
    wait_async_lds();
    __syncthreads();

    // ---- energies (energy_H has the masked diagonal, energy_W does not) ----
    for (int i = tid; i < 16 * 96; i += 32) {
        int hh = i / 96, j = i % 96;
        float e = qt[0][hh] * kcol[0][j] + qt[1][hh] * kcol[1][j];
        if (j == h0 + hh) e = NEG_INF_;
        attH[hh][j] = e;
    }
    for (int i = tid; i < 16 * 192; i += 32) {
        int hh = i / 192, j = i % 192;
        attW[hh][j] = qt[0][hh] * krow[0][hh][j] + qt[1][hh] * krow[1][hh][j];
    }
    __syncthreads();

    // ---- joint softmax over [attH(96) ++ attW(192)] per row hh ----
    for (int hh = 0; hh < 16; ++hh) {
        float mx = -3.0e38f;
        for (int i = tid; i < 288; i += 32) {
            float e = (i < 96) ? attH[hh][i] : attW[hh][i - 96];
            mx = fmaxf(mx, e);
        }
#pragma unroll
        for (int off = 16; off; off >>= 1) mx = fmaxf(mx, __shfl_xor(mx, off, 32));
        float sum = 0.f;
        for (int i = tid; i < 288; i += 32) {
            float e = __expf(((i < 96) ? attH[hh][i] : attW[hh][i - 96]) - mx);
            if (i < 96) attH[hh][i] = e; else attW[hh][i - 96] = e;
            sum += e;
        }
#pragma unroll
        for (int off = 16; off; off >>= 1) sum += __shfl_xor(sum, off, 32);
        float inv = 1.0f / sum;
        for (int i = tid; i < 288; i += 32) {
            if (i < 96) attH[hh][i] *= inv; else attW[hh][i - 96] *= inv;
        }
    }
    __syncthreads();

    // ---- out_H: D[hh][c] = sum_j attH[hh][j] * vcol[c][j] : 16x16x96 GEMM ----
    v8f oh = {0.f, 0.f, 0.f, 0.f, 0.f, 0.f, 0.f, 0.f};
#pragma unroll
    for (int kb4 = 0; kb4 < 96; kb4 += 4) {
        v2f a, bm;
        a[0]  = attH[m][kb4 + 2 * hi];
        a[1]  = attH[m][kb4 + 2 * hi + 1];
        bm[0] = vcol[m][kb4 + 2 * hi];
        bm[1] = vcol[m][kb4 + 2 * hi + 1];
        oh = __builtin_amdgcn_wmma_f32_16x16x4_f32(
            false, a, false, bm, (short)0, oh, false, false);
    }

    // ---- out_W: batched matvec (v row depends on hh). Async-stage each row,
    // split the 192-wide dot over half-waves, combine with cross-half shfl,
    // park results in the same (reg r, half hi) slots as the WMMA accumulator.
    float ow[8];
#pragma unroll
    for (int r = 0; r < 8; ++r) ow[r] = 0.f;
#pragma unroll
    for (int hh = 0; hh < 16; ++hh) {
        for (int i = tid; i < 16 * 48; i += 32) {   // 16 rows x 48 B128 chunks
            int c = i / 48, j = (i % 48) * 4;
            async_ld_b128(&vr[c][j], &v[((b * C_ + c) * H_ + h0 + hh) * W_ + j]);
        }
        wait_async_lds();
        __syncthreads();
        float part = 0.f;
        int c = m, jb = hi * 96;
        for (int j = 0; j < 96; ++j)
            part += attW[hh][jb + j] * vr[c][jb + j];
        part += __shfl_xor(part, 16, 32);
        if ((hh >> 3) == hi) ow[hh & 7] = part;
        __syncthreads();
    }

    // ---- residual + gamma, layout-flagged store (folds the transpose) ----
    float g = gamma[0];
#pragma unroll
    for (int r = 0; r < 8; ++r) {
        int hA = h0 + r + 8 * hi;
        int c  = m;
        int xin  = in_cdhw  ? ((c * D_ + b) * H_ + hA) * W_ + w
                            : ((b * C_ + c) * H_ + hA) * W_ + w;
        int oidx = out_cdhw ? ((c * D_ + b) * H_ + hA) * W_ + w
                            : ((b * C_ + c) * H_ + hA) * W_ + w;
        out[oidx] = g * (oh[r] + ow[r]) + x[xin];
    }
}

// ---------------------------------------------------------------------------
extern "C" void kernel_launch(void* const* d_in, const int* in_sizes, int n_in,
                              void* d_out, int out_size, void* d_ws, size_t ws_size,
                              hipStream_t stream) {
    const float* cv = (const float*)d_in[0];  // (1,16,24,96,192) f32
    const float* qw = (const float*)d_in[1];
    const float* qb = (const float*)d_in[2];
    const float* kw = (const float*)d_in[3];
    const float* kb = (const float*)d_in[4];
    const float* vw = (const float*)d_in[5];
    const float* vb = (const float*)d_in[6];
    const float* gm = (const float*)d_in[7];
    float* out = (float*)d_out;
    float* ws  = (float*)d_ws;

    const size_t SZ_QK = (size_t)D_ * 2 * H_ * W_;   // 884,736 f
    const size_t SZ_X  = (size_t)D_ * C_ * H_ * W_;  // 7,077,888 f
    float* qbuf = ws;
    float* kbuf = ws + SZ_QK;
    float* vbuf = ws + 2 * SZ_QK;
    float* x1   = ws + 2 * SZ_QK + SZ_X;             // inter-pass tensor (dchw)

    dim3 blk(32);
    dim3 gproj(W_ / 16, H_, D_);
    dim3 gatt(W_, H_ / 16, D_);

    // pass 1: read cost_volume in (c,d,h,w) layout, write x1 in (d,c,h,w)
    cca_proj_kernel<<<gproj, blk, 0, stream>>>(cv, qw, qb, kw, kb, vw, vb,
                                               qbuf, kbuf, vbuf, 1);
    cca_attend_kernel<<<gatt, blk, 0, stream>>>(cv, qbuf, kbuf, vbuf, gm, x1, 1, 0);

    // pass 2: read x1 (d,c,h,w), write final output transposed back to (c,d,h,w)
    cca_proj_kernel<<<gproj, blk, 0, stream>>>(x1, qw, qb, kw, kb, vw, vb,
                                               qbuf, kbuf, vbuf, 0);
    cca_attend_kernel<<<gatt, blk, 0, stream>>>(x1, qbuf, kbuf, vbuf, gm, out, 0, 1);
}